// eGAUSSp_80822694576472
// MI455X (gfx1250) — compile-verified
//
#include <hip/hip_runtime.h>

// ---------------------------------------------------------------------------
// eGAUSSp inference on MI455X (gfx1250, wave32)
//
//   d2[b,c] = (x_b - mu_c)^T Ainv_c (x_b - mu_c)
//           = sum_ij x_i x_j Ainv[i,j]  +  x . (-2 Ainv mu)  +  mu^T Ainv mu
//
// => one fp32 WMMA GEMM:  X2[2048 x 1056]  @  B[1056 x 1024]  (+ bias s[c])
//    X2[b, i*32+j] = x_i*x_j   (k <  1024)
//    X2[b, 1024+i] = x_i       (k >= 1024)
//    B[i*32+j, c] = Ainv_c[i,j] ; B[1024+i, c] = -2*(Ainv_c mu_c)_i
//    stored c-major as BTT[c][k] so async LDS staging is contiguous.
// ---------------------------------------------------------------------------

typedef __attribute__((ext_vector_type(2))) float v2f;
typedef __attribute__((ext_vector_type(8))) float v8f;

#define BDIM 2048
#define CDIM 1024
#define DDIM 32
#define KDIM 1056        // 32*32 + 32
#define NCLS 10

// ---------------- workspace layout (bytes) ----------------
#define WS_BTT    0u                                   // [1024][1056] f32 = 4,325,376
#define WS_SARR   4325376u                             // [1024] f32
#define WS_LABELS 4329472u                             // [1024] i32
#define WS_X2     4333568u                             // [2048][1056] f32 = 8,650,752
#define WS_GAMMA  12984320u                            // [2048][1024] f32 = 8,388,608
// total ~21.4 MB

// ---------------------------------------------------------------------------
// CDNA5 async memory->LDS copy (16B per lane), tracked by ASYNCcnt.
// The LDS byte address is the low 32 bits of the generic pointer
// (ISA 10.2: flat->LDS uses addr[31:0]).
// ---------------------------------------------------------------------------
__device__ __forceinline__ void async_load_b128(unsigned lds_addr, const void* gptr) {
    asm volatile("global_load_async_to_lds_b128 %0, %1, off"
                 :: "v"(lds_addr),
                    "v"((unsigned long long)(size_t)gptr)
                 : "memory");
}
__device__ __forceinline__ void wait_asynccnt0() {
    asm volatile("s_wait_asynccnt 0x0" ::: "memory");
}

// ---------------------------------------------------------------------------
// Kernel 1: per-cluster Gauss-Jordan inversion of cov = S/n (fallback 0.1*I),
// emit BTT[c][k], bias s[c] = mu^T Ainv mu, label[c].
// ---------------------------------------------------------------------------
__global__ __launch_bounds__(256) void invert_cov_kernel(
    const float* __restrict__ S, const float* __restrict__ mu,
    const float* __restrict__ n, const int* __restrict__ onehot,
    float* __restrict__ BTT, float* __restrict__ s_arr, int* __restrict__ labels)
{
    __shared__ float M[32 * 66];   // 64 cols + 2 pad
    __shared__ float mu_sh[32];
    __shared__ float amu_sh[32];
    __shared__ float pivinv;

    const int c   = blockIdx.x;
    const int tid = threadIdx.x;
    const int r   = tid >> 3;            // 0..31 : my row
    const int c0  = (tid & 7) * 8;       // my 8-column slice of the 64 cols

    const float nc     = n[c];
    const bool  mature = nc > 1.0f;
    const float invn   = mature ? (1.0f / nc) : 1.0f;

    if (tid < 32) mu_sh[tid] = mu[c * DDIM + tid];

    #pragma unroll
    for (int j = 0; j < 8; ++j) {
        const int col = c0 + j;
        float v;
        if (col < 32) {
            v = mature ? S[(size_t)c * (DDIM * DDIM) + r * DDIM + col] * invn
                       : ((col == r) ? 0.1f : 0.0f);
        } else {
            v = ((col - 32) == r) ? 1.0f : 0.0f;
        }
        M[r * 66 + col] = v;
    }
    __syncthreads();

    // Gauss-Jordan (SPD: no pivoting needed)
    for (int k = 0; k < 32; ++k) {
        if (tid == 0) pivinv = 1.0f / M[k * 66 + k];
        __syncthreads();
        if (r == k) {
            #pragma unroll
            for (int j = 0; j < 8; ++j) M[k * 66 + c0 + j] *= pivinv;
        }
        __syncthreads();
        const float f = (r == k) ? 0.0f : M[r * 66 + k];
        __syncthreads();
        #pragma unroll
        for (int j = 0; j < 8; ++j)
            M[r * 66 + c0 + j] -= f * M[k * 66 + c0 + j];
        __syncthreads();
    }

    // write inverse rows, c-major: BTT[c*KDIM + (i*32+j)]
    if (c0 >= 32) {
        #pragma unroll
        for (int j = 0; j < 8; ++j) {
            const int jj = c0 - 32 + j;
            BTT[(size_t)c * KDIM + (r * 32 + jj)] = M[r * 66 + 32 + jj];
        }
    }
    if (tid < 32) {
        float a = 0.0f;
        #pragma unroll
        for (int j = 0; j < 32; ++j) a += M[tid * 66 + 32 + j] * mu_sh[j];
        amu_sh[tid] = a;
        BTT[(size_t)c * KDIM + 1024 + tid] = -2.0f * a;
    }
    __syncthreads();
    if (tid == 0) {
        float s = 0.0f;
        #pragma unroll
        for (int i = 0; i < 32; ++i) s += mu_sh[i] * amu_sh[i];
        s_arr[c] = s;
        int lab = 0;
        for (int k = 0; k < NCLS; ++k)
            if (onehot[c * NCLS + k] != 0) { lab = k; break; }
        labels[c] = lab;
    }
}

// ---------------------------------------------------------------------------
// Kernel 2: lifted features X2[b, :]
// ---------------------------------------------------------------------------
__global__ __launch_bounds__(256) void build_x2_kernel(
    const float* __restrict__ data, float* __restrict__ X2)
{
    __shared__ float x[32];
    const int b = blockIdx.x;
    if (threadIdx.x < 32) x[threadIdx.x] = data[b * DDIM + threadIdx.x];
    __syncthreads();
    for (int k = threadIdx.x; k < KDIM; k += 256) {
        float v;
        if (k < 1024) v = x[k >> 5] * x[k & 31];
        else          v = x[k - 1024];
        X2[(size_t)b * KDIM + k] = v;
    }
}

// ---------------------------------------------------------------------------
// Kernel 3: fp32 WMMA GEMM [2048 x 1056] @ [1056 x 1024] + fused exp epilogue.
// Block = 256 thr (8 waves, 4x2); block tile 128 x 64; wave tile 32 x 32
// (2x2 register blocking -> 4 accumulators, 4 wmma per 2A+2B fragment loads).
// Staging uses CDNA5 GLOBAL_LOAD_ASYNC_TO_LDS_B128 (no VGPR round-trip),
// waited with S_WAIT_ASYNCCNT. LDS row stride 36 floats: every 16B async
// destination is 16B-aligned and fragment ds_load_b64's are bank-conflict-free.
// ---------------------------------------------------------------------------
#define TILE_M 128
#define TILE_N 64
#define LDSTR  36

__global__ __launch_bounds__(256) void gemm_gamma_kernel(
    const float* __restrict__ X2, const float* __restrict__ BTT,
    const float* __restrict__ s_arr, float* __restrict__ gamma)
{
    __shared__ __align__(16) float Alds[TILE_M * LDSTR];   // [row][k]
    __shared__ __align__(16) float BldsT[TILE_N * LDSTR];  // [col][k]

    const int tid  = threadIdx.x;
    const int lane = tid & 31;
    const int wave = tid >> 5;           // 0..7
    const int half = lane >> 4;          // 0/1
    const int l16  = lane & 15;

    const int rowBase = blockIdx.x * TILE_M;     // batch rows
    const int colBase = blockIdx.y * TILE_N;     // cluster cols
    const int wr = (wave & 3) * 32;              // wave row inside block tile
    const int wc = (wave >> 2) * 32;             // wave col inside block tile

    // LDS byte offsets (low 32 bits of generic pointer = LDS address)
    const unsigned aBase = (unsigned)(size_t)(void*)Alds;
    const unsigned bBase = (unsigned)(size_t)(void*)BldsT;

    v8f acc00 = {}, acc01 = {}, acc10 = {}, acc11 = {};

    for (int kc = 0; kc < KDIM; kc += 32) {
        __syncthreads();   // all waves done reading previous chunk

        // A chunk: 128 rows x 32 K -> 1024 x16B async copies, 4 per thread
        #pragma unroll
        for (int it = 0; it < 4; ++it) {
            const int idx = tid + it * 256;
            const int r   = idx >> 3;
            const int cc  = (idx & 7) * 4;
            async_load_b128(aBase + (unsigned)(r * LDSTR + cc) * 4u,
                            &X2[(size_t)(rowBase + r) * KDIM + kc + cc]);
        }
        // B chunk: 64 cols x 32 K -> 512 x16B async copies, 2 per thread
        #pragma unroll
        for (int it = 0; it < 2; ++it) {
            const int idx = tid + it * 256;
            const int col = idx >> 3;        // 0..63
            const int cc  = (idx & 7) * 4;   // 0..28
            async_load_b128(bBase + (unsigned)(col * LDSTR + cc) * 4u,
                            &BTT[(size_t)(colBase + col) * KDIM + kc + cc]);
        }
        wait_asynccnt0();
        __syncthreads();   // LDS writes visible to all waves

        // 8 K-steps x 4 wmma = 32 x V_WMMA_F32_16X16X4_F32 per chunk
        #pragma unroll
        for (int k0 = 0; k0 < 32; k0 += 4) {
            const int ko = k0 + 2 * half;    // K = 2*half + vgpr (A & B layouts)
            const v2f a0 = *reinterpret_cast<const v2f*>(&Alds[(wr + l16) * LDSTR + ko]);
            const v2f a1 = *reinterpret_cast<const v2f*>(&Alds[(wr + 16 + l16) * LDSTR + ko]);
            const v2f b0 = *reinterpret_cast<const v2f*>(&BldsT[(wc + l16) * LDSTR + ko]);
            const v2f b1 = *reinterpret_cast<const v2f*>(&BldsT[(wc + 16 + l16) * LDSTR + ko]);
            acc00 = __builtin_amdgcn_wmma_f32_16x16x4_f32(false, a0, false, b0, (short)0, acc00, false, false);
            acc01 = __builtin_amdgcn_wmma_f32_16x16x4_f32(false, a0, false, b1, (short)0, acc01, false, false);
            acc10 = __builtin_amdgcn_wmma_f32_16x16x4_f32(false, a1, false, b0, (short)0, acc10, false, false);
            acc11 = __builtin_amdgcn_wmma_f32_16x16x4_f32(false, a1, false, b1, (short)0, acc11, false, false);
        }
    }

    // epilogue: d2 = acc + s[col]; gamma = exp(-d2/32)
    const int   col0 = colBase + wc + l16;
    const int   col1 = col0 + 16;
    const float sc0  = s_arr[col0];
    const float sc1  = s_arr[col1];
    #pragma unroll
    for (int r = 0; r < 8; ++r) {
        const int row0 = rowBase + wr + r + 8 * half;   // C/D layout: M = r + 8*half
        const int row1 = row0 + 16;
        gamma[(size_t)row0 * CDIM + col0] = __expf(-(acc00[r] + sc0) * (1.0f / 32.0f));
        gamma[(size_t)row0 * CDIM + col1] = __expf(-(acc01[r] + sc1) * (1.0f / 32.0f));
        gamma[(size_t)row1 * CDIM + col0] = __expf(-(acc10[r] + sc0) * (1.0f / 32.0f));
        gamma[(size_t)row1 * CDIM + col1] = __expf(-(acc11[r] + sc1) * (1.0f / 32.0f));
    }
}

// ---------------------------------------------------------------------------
// Kernel 4: per-row normalize / label vote / argmax. One block per batch row.
// out = [2048x10 label_scores][2048 preds][2048 clusters]   (all as f32)
// ---------------------------------------------------------------------------
__global__ __launch_bounds__(256) void reduce_kernel(
    const float* __restrict__ gamma, const int* __restrict__ labels,
    float* __restrict__ out)
{
    __shared__ float ssum;
    __shared__ float sclass[NCLS];
    __shared__ unsigned long long smax;

    const int b   = blockIdx.x;
    const int tid = threadIdx.x;
    if (tid == 0) { ssum = 0.0f; smax = 0ull; }
    if (tid < NCLS) sclass[tid] = 0.0f;
    __syncthreads();

    float lsum = 0.0f;
    unsigned long long lpack = 0ull;
    for (int c = tid; c < CDIM; c += 256) {
        const float g = gamma[(size_t)b * CDIM + c];
        lsum += g;
        atomicAdd(&sclass[labels[c]], g);
        // gamma >= 0 -> float bits are order-preserving; inverted index gives
        // first-occurrence tie-break like jnp.argmax.
        const unsigned long long p =
            ((unsigned long long)__float_as_uint(g) << 32) |
            (unsigned long long)(0xFFFFFFFFu - (unsigned)c);
        lpack = (p > lpack) ? p : lpack;
    }
    atomicAdd(&ssum, lsum);
    atomicMax(&smax, lpack);
    __syncthreads();

    if (tid == 0) {
        const float inv = 1.0f / fmaxf(ssum, 1e-30f);
        int best = 0; float bv = -1.0f;
        #pragma unroll
        for (int k = 0; k < NCLS; ++k) {
            const float v = sclass[k] * inv;
            out[b * NCLS + k] = v;
            if (v > bv) { bv = v; best = k; }
        }
        out[BDIM * NCLS + b]        = (float)best;
        out[BDIM * NCLS + BDIM + b] =
            (float)(0xFFFFFFFFu - (unsigned)(smax & 0xFFFFFFFFull));
    }
}

// ---------------------------------------------------------------------------
extern "C" void kernel_launch(void* const* d_in, const int* in_sizes, int n_in,
                              void* d_out, int out_size, void* d_ws, size_t ws_size,
                              hipStream_t stream)
{
    const float* data   = (const float*)d_in[0];   // [2048,32]
    const float* mu     = (const float*)d_in[1];   // [1024,32]
    const float* S      = (const float*)d_in[2];   // [1024,32,32]
    const float* n      = (const float*)d_in[3];   // [1024]
    const int*   onehot = (const int*)d_in[4];     // [1024,10]
    float* out = (float*)d_out;

    char* ws = (char*)d_ws;
    float* BTT    = (float*)(ws + WS_BTT);
    float* s_arr  = (float*)(ws + WS_SARR);
    int*   labels = (int*)(ws + WS_LABELS);
    float* X2     = (float*)(ws + WS_X2);
    float* gamma  = (float*)(ws + WS_GAMMA);

    invert_cov_kernel<<<CDIM, 256, 0, stream>>>(S, mu, n, onehot, BTT, s_arr, labels);
    build_x2_kernel<<<BDIM, 256, 0, stream>>>(data, X2);
    gemm_gamma_kernel<<<dim3(BDIM / TILE_M, CDIM / TILE_N), 256, 0, stream>>>(X2, BTT, s_arr, gamma);
    reduce_kernel<<<BDIM, 256, 0, stream>>>(gamma, labels, out);
}